// shapenet_PVCNN_16733192585418
// MI455X (gfx1250) — compile-verified
//
#include <hip/hip_runtime.h>
#include <hip/hip_bf16.h>
#include <math.h>

// ---------------------------------------------------------------------------
// PVCNN-style point classifier for MI455X (gfx1250, wave32).
// GEMMs: V_WMMA_F32_16X16X4_F32, 16x64 output tile per wave (4 WMMAs / K-step),
// compile-time CI, zero-padded weights => no predicated loads in inner loop.
// ---------------------------------------------------------------------------

#define NPTS  2048
#define BATCH 4
#define COLS  (BATCH * NPTS)   // 8192 fused (batch, n) columns
#define EPSF  1e-5f

typedef __attribute__((ext_vector_type(2))) float v2f;
typedef __attribute__((ext_vector_type(8))) float v8f;

__device__ __forceinline__ float angle3(float ux, float uy, float uz,
                                        float vx, float vy, float vz) {
  float cx = uy * vz - uz * vy;
  float cy = uz * vx - ux * vz;
  float cz = ux * vy - uy * vx;
  float s  = sqrtf(cx * cx + cy * cy + cz * cz);
  float c  = ux * vx + uy * vy + uz * vz;
  return atan2f(s, c);
}

// ---------------------------------------------------------------------------
// 1) Per-batch centroids + normalized per-point normals.
// cstat per batch: [cc(3), cn(3), ncn(3)]
// ---------------------------------------------------------------------------
__global__ void centroid_kernel(const float* __restrict__ in,
                                float* __restrict__ nrm,
                                float* __restrict__ cstat) {
  int b = blockIdx.x;
  int tid = threadIdx.x;
  const float* base = in + (size_t)b * 22 * NPTS;
  float sc0 = 0.f, sc1 = 0.f, sc2 = 0.f;
  float sn0 = 0.f, sn1 = 0.f, sn2 = 0.f;
  for (int n = tid; n < NPTS; n += 256) {
    float px = base[0 * NPTS + n], py = base[1 * NPTS + n], pz = base[2 * NPTS + n];
    float nx = base[3 * NPTS + n], ny = base[4 * NPTS + n], nz = base[5 * NPTS + n];
    float il = rsqrtf(nx * nx + ny * ny + nz * nz);
    nx *= il; ny *= il; nz *= il;
    size_t pi = ((size_t)b * NPTS + n) * 3;
    nrm[pi + 0] = nx; nrm[pi + 1] = ny; nrm[pi + 2] = nz;
    sc0 += px; sc1 += py; sc2 += pz;
    sn0 += nx; sn1 += ny; sn2 += nz;
  }
  __shared__ float red[6][256];
  red[0][tid] = sc0; red[1][tid] = sc1; red[2][tid] = sc2;
  red[3][tid] = sn0; red[4][tid] = sn1; red[5][tid] = sn2;
  __syncthreads();
  for (int s = 128; s > 0; s >>= 1) {
    if (tid < s)
      for (int q = 0; q < 6; ++q) red[q][tid] += red[q][tid + s];
    __syncthreads();
  }
  if (tid == 0) {
    float inv = 1.0f / (float)NPTS;
    float cc0 = red[0][0] * inv, cc1 = red[1][0] * inv, cc2 = red[2][0] * inv;
    float cn0 = red[3][0] * inv, cn1 = red[4][0] * inv, cn2 = red[5][0] * inv;
    float il = rsqrtf(cn0 * cn0 + cn1 * cn1 + cn2 * cn2);
    float* cs = cstat + b * 9;
    cs[0] = cc0; cs[1] = cc1; cs[2] = cc2;
    cs[3] = cn0; cs[4] = cn1; cs[5] = cn2;
    cs[6] = cn0 * il; cs[7] = cn1 * il; cs[8] = cn2 * il;
  }
}

// ---------------------------------------------------------------------------
// 2) PPF features (feats rows 0..3), projection vectors P, zero rows 5..7.
// ---------------------------------------------------------------------------
__global__ void ppf_kernel(const float* __restrict__ in,
                           const float* __restrict__ nrm,
                           const float* __restrict__ cstat,
                           float* __restrict__ feats,
                           float* __restrict__ P) {
  int idx = blockIdx.x * 256 + threadIdx.x;
  if (idx >= COLS) return;
  int b = idx >> 11, n = idx & (NPTS - 1);
  const float* base = in + (size_t)b * 22 * NPTS;
  float px = base[0 * NPTS + n], py = base[1 * NPTS + n], pz = base[2 * NPTS + n];
  const float* cs = cstat + b * 9;
  float ccx = cs[0], ccy = cs[1], ccz = cs[2];
  float cnx = cs[3], cny = cs[4], cnz = cs[5];
  float nnx = cs[6], nny = cs[7], nnz = cs[8];
  size_t pi = (size_t)idx * 3;
  float nx = nrm[pi], ny = nrm[pi + 1], nz = nrm[pi + 2];
  float dx = px - ccx, dy = py - ccy, dz = pz - ccz;
  feats[0 * COLS + idx] = angle3(cnx, cny, cnz, dx, dy, dz);
  feats[1 * COLS + idx] = angle3(nx, ny, nz, dx, dy, dz);
  feats[2 * COLS + idx] = angle3(cnx, cny, cnz, nx, ny, nz);
  feats[3 * COLS + idx] = sqrtf(dx * dx + dy * dy + dz * dz);
  float dot = dx * nnx + dy * nny + dz * nnz;   // dot vs ncn, subtract cn (ref)
  P[pi + 0] = dx - dot * cnx;
  P[pi + 1] = dy - dot * cny;
  P[pi + 2] = dz - dot * cnz;
  feats[5 * COLS + idx] = 0.f;
  feats[6 * COLS + idx] = 0.f;
  feats[7 * COLS + idx] = 0.f;
}

// ---------------------------------------------------------------------------
// 3) alpha: exact per-row median of 2048 pairwise angles via radix select
//    on float bits (all values > 0 => uint order == float order).
// ---------------------------------------------------------------------------
__global__ void alpha_kernel(const float* __restrict__ P,
                             float* __restrict__ feats) {
  int row = blockIdx.x;          // b*NPTS + i
  int b = row >> 11;
  int tid = threadIdx.x;

  __shared__ unsigned bits[NPTS];
  __shared__ unsigned hist[256];
  __shared__ float Pi[3];
  __shared__ unsigned s_prefix;
  __shared__ int s_k;

  if (tid < 3) Pi[tid] = P[(size_t)row * 3 + tid];
  if (tid == 0) { s_prefix = 0u; s_k = (NPTS - 1) >> 1; }  // k = 1023
  __syncthreads();
  float ix = Pi[0], iy = Pi[1], iz = Pi[2];
  const float* Pb = P + (size_t)b * NPTS * 3;
  for (int j = tid; j < NPTS; j += 256) {
    float jx = Pb[j * 3], jy = Pb[j * 3 + 1], jz = Pb[j * 3 + 2];
    float cx = iy * jz - iz * jy;
    float cy = iz * jx - ix * jz;
    float cz = ix * jy - iy * jx;
    float s = sqrtf(cx * cx + cy * cy + cz * cz);
    float c = ix * jx + iy * jy + iz * jz;
    float at = atan2f(s, c);
    if (at <= 1e-5f) at = 100.0f;
    bits[j] = __float_as_uint(at);
  }
  __syncthreads();

  for (int pass = 3; pass >= 0; --pass) {
    int shift = pass * 8;
    unsigned pmask = (pass == 3) ? 0u : (0xFFFFFFFFu << (shift + 8));
    hist[tid] = 0u;
    __syncthreads();
    unsigned pfx = s_prefix;
    for (int j = tid; j < NPTS; j += 256) {
      unsigned v = bits[j];
      if ((v & pmask) == pfx) atomicAdd(&hist[(v >> shift) & 255u], 1u);
    }
    __syncthreads();
    if (tid == 0) {
      int k = s_k, acc = 0;
      unsigned sel = 255u;
      for (int q = 0; q < 256; ++q) {
        int c2 = (int)hist[q];
        if (acc + c2 > k) { sel = (unsigned)q; break; }
        acc += c2;
      }
      s_prefix = pfx | (sel << shift);
      s_k = k - acc;
    }
    __syncthreads();
  }
  if (tid == 0) feats[4 * COLS + row] = __uint_as_float(s_prefix);
}

// ---------------------------------------------------------------------------
// 4) one-hot rows -> xcat rows 0..15
// ---------------------------------------------------------------------------
__global__ void onehot_kernel(const float* __restrict__ in,
                              float* __restrict__ xcat) {
  size_t idx = (size_t)blockIdx.x * 256 + threadIdx.x;
  if (idx >= (size_t)16 * COLS) return;
  int c = (int)(idx / COLS);
  int col = (int)(idx % COLS);
  int b = col >> 11, n = col & (NPTS - 1);
  xcat[idx] = in[((size_t)b * 22 + 6 + c) * NPTS + n];
}

// ---------------------------------------------------------------------------
// 5) Zero-pad a weight matrix (co x ci) -> (co_pad x ci_pad)
// ---------------------------------------------------------------------------
__global__ void pad_weight_kernel(const float* __restrict__ src,
                                  float* __restrict__ dst,
                                  int co, int ci, int co_pad, int ci_pad) {
  int idx = blockIdx.x * 256 + threadIdx.x;
  if (idx >= co_pad * ci_pad) return;
  int r = idx / ci_pad, c = idx % ci_pad;
  dst[idx] = (r < co && c < ci) ? src[r * ci + c] : 0.f;
}

// ---------------------------------------------------------------------------
// 6) WMMA fp32 GEMM: Y(co x COLS) = W(co x CI) * X(CI x COLS).
//    CI compile-time (multiple of 4). One wave -> 16x64 tile: one A fragment
//    per K-step feeds 4 WMMAs. No guards => unconditional clause-able loads.
//    Fragment mapping (wave32, V_WMMA_F32_16X16X4_F32):
//      A (16x4): lane m = lane&15, vgpr v holds K = 2*(lane>=16)+v
//      B (4x16): lane n = lane&15, vgpr v holds K = 2*(lane>=16)+v
//      C/D     : lane&15 = N, vgpr r -> M = r + 8*(lane>=16)
// ---------------------------------------------------------------------------
template <int CI>
__global__ void wmma_gemm_t(const float* __restrict__ W,
                            const float* __restrict__ X,
                            float* __restrict__ Y, int co) {
  const int nBlocks = COLS / 64;   // 128 tiles of width 64
  int wave = threadIdx.x >> 5;
  int tile = blockIdx.x * (blockDim.x >> 5) + wave;
  int total = (co >> 4) * nBlocks;
  if (tile >= total) return;
  int oT = tile / nBlocks, nT = tile % nBlocks;
  int lane = threadIdx.x & 31;
  int m = lane & 15;
  int hi = lane >> 4;
  int col0 = nT * 64 + m;
  const float* Wrow = W + (size_t)(oT * 16 + m) * CI + hi * 2;
  const float* Xcol = X + col0;

  v8f acc0 = {}, acc1 = {}, acc2 = {}, acc3 = {};
#pragma unroll 4
  for (int k0 = 0; k0 < CI; k0 += 4) {
    v2f a;
    a.x = Wrow[k0];
    a.y = Wrow[k0 + 1];
    const float* xp = Xcol + (size_t)(k0 + hi * 2) * COLS;
    if (CI > 64 && (k0 & 15) == 0)
      __builtin_prefetch(xp + (size_t)16 * COLS, 0, 0);
    v2f b0, b1, b2, b3;
    b0.x = xp[0];  b0.y = xp[COLS + 0];
    b1.x = xp[16]; b1.y = xp[COLS + 16];
    b2.x = xp[32]; b2.y = xp[COLS + 32];
    b3.x = xp[48]; b3.y = xp[COLS + 48];
    acc0 = __builtin_amdgcn_wmma_f32_16x16x4_f32(false, a, false, b0, (short)0, acc0, false, false);
    acc1 = __builtin_amdgcn_wmma_f32_16x16x4_f32(false, a, false, b1, (short)0, acc1, false, false);
    acc2 = __builtin_amdgcn_wmma_f32_16x16x4_f32(false, a, false, b2, (short)0, acc2, false, false);
    acc3 = __builtin_amdgcn_wmma_f32_16x16x4_f32(false, a, false, b3, (short)0, acc3, false, false);
  }
  int orow = oT * 16 + hi * 8;
  float* Yb = Y + (size_t)orow * COLS + col0;
#pragma unroll
  for (int r = 0; r < 8; ++r) {
    Yb[(size_t)r * COLS + 0]  = acc0[r];
    Yb[(size_t)r * COLS + 16] = acc1[r];
    Yb[(size_t)r * COLS + 32] = acc2[r];
    Yb[(size_t)r * COLS + 48] = acc3[r];
  }
}

// ---------------------------------------------------------------------------
// 7) BN statistics per output channel
// ---------------------------------------------------------------------------
__global__ void bn_stats_kernel(const float* __restrict__ Y,
                                float* __restrict__ mu,
                                float* __restrict__ var) {
  int c = blockIdx.x;
  int tid = threadIdx.x;
  const float* row = Y + (size_t)c * COLS;
  float s = 0.f, sq = 0.f;
  for (int j = tid; j < COLS; j += 256) {
    float v = row[j];
    s += v; sq += v * v;
  }
  __shared__ float rs[256], rq[256];
  rs[tid] = s; rq[tid] = sq;
  __syncthreads();
  for (int st = 128; st > 0; st >>= 1) {
    if (tid < st) { rs[tid] += rs[tid + st]; rq[tid] += rq[tid + st]; }
    __syncthreads();
  }
  if (tid == 0) {
    float mean = rs[0] / (float)COLS;
    mu[c] = mean;
    var[c] = rq[0] / (float)COLS - mean * mean;
  }
}

// ---------------------------------------------------------------------------
// 8) BN affine + ReLU
// ---------------------------------------------------------------------------
__global__ void bn_apply_relu(const float* __restrict__ Y,
                              const float* __restrict__ mu,
                              const float* __restrict__ var,
                              const float* __restrict__ g,
                              const float* __restrict__ bia,
                              float* __restrict__ dst, int co) {
  size_t idx = (size_t)blockIdx.x * 256 + threadIdx.x;
  if (idx >= (size_t)co * COLS) return;
  int c = (int)(idx / COLS);
  float v = (Y[idx] - mu[c]) * rsqrtf(var[c] + EPSF) * g[c] + bia[c];
  dst[idx] = v > 0.f ? v : 0.f;
}

// ---------------------------------------------------------------------------
// 9) Max over N per (channel, batch), broadcast over N
// ---------------------------------------------------------------------------
__global__ void maxpool_bcast(const float* __restrict__ f4,
                              float* __restrict__ dst) {
  int c = blockIdx.x >> 2;
  int b = blockIdx.x & 3;
  int tid = threadIdx.x;
  const float* row = f4 + (size_t)c * COLS + b * NPTS;
  float m = -3.402823466e38f;
  for (int j = tid; j < NPTS; j += 256) m = fmaxf(m, row[j]);
  __shared__ float r[256];
  r[tid] = m;
  __syncthreads();
  for (int s = 128; s > 0; s >>= 1) {
    if (tid < s) r[tid] = fmaxf(r[tid], r[tid + s]);
    __syncthreads();
  }
  float mv = r[0];
  float* drow = dst + (size_t)c * COLS + b * NPTS;
  for (int j = tid; j < NPTS; j += 256) drow[j] = mv;
}

// ---------------------------------------------------------------------------
// 10) Final 128->50 GEMM (+bias) on padded 64x128 weights; store mask row<50,
//     output layout (B, 50, N).
// ---------------------------------------------------------------------------
__global__ void wmma_gemm_final(const float* __restrict__ Wp,   // 64 x 128
                                const float* __restrict__ biasp, // 64
                                const float* __restrict__ X,
                                float* __restrict__ out) {
  constexpr int CI = 128;
  const int nBlocks = COLS / 64;   // 128
  int wave = threadIdx.x >> 5;
  int tile = blockIdx.x * (blockDim.x >> 5) + wave;
  int total = 4 * nBlocks;         // co_pad/16 = 4
  if (tile >= total) return;
  int oT = tile / nBlocks, nT = tile % nBlocks;
  int lane = threadIdx.x & 31;
  int m = lane & 15;
  int hi = lane >> 4;
  int col0 = nT * 64 + m;
  const float* Wrow = Wp + (size_t)(oT * 16 + m) * CI + hi * 2;
  const float* Xcol = X + col0;

  v8f acc0 = {}, acc1 = {}, acc2 = {}, acc3 = {};
#pragma unroll 4
  for (int k0 = 0; k0 < CI; k0 += 4) {
    v2f a;
    a.x = Wrow[k0];
    a.y = Wrow[k0 + 1];
    const float* xp = Xcol + (size_t)(k0 + hi * 2) * COLS;
    v2f b0, b1, b2, b3;
    b0.x = xp[0];  b0.y = xp[COLS + 0];
    b1.x = xp[16]; b1.y = xp[COLS + 16];
    b2.x = xp[32]; b2.y = xp[COLS + 32];
    b3.x = xp[48]; b3.y = xp[COLS + 48];
    acc0 = __builtin_amdgcn_wmma_f32_16x16x4_f32(false, a, false, b0, (short)0, acc0, false, false);
    acc1 = __builtin_amdgcn_wmma_f32_16x16x4_f32(false, a, false, b1, (short)0, acc1, false, false);
    acc2 = __builtin_amdgcn_wmma_f32_16x16x4_f32(false, a, false, b2, (short)0, acc2, false, false);
    acc3 = __builtin_amdgcn_wmma_f32_16x16x4_f32(false, a, false, b3, (short)0, acc3, false, false);
  }
  int bcol = col0 >> 11, nc = col0 & (NPTS - 1);
#pragma unroll
  for (int r = 0; r < 8; ++r) {
    int row = oT * 16 + hi * 8 + r;
    if (row < 50) {
      float bb = biasp[row];
      float* o = out + ((size_t)bcol * 50 + row) * NPTS + nc;
      o[0]  = acc0[r] + bb;
      o[16] = acc1[r] + bb;
      o[32] = acc2[r] + bb;
      o[48] = acc3[r] + bb;
    }
  }
}

// ---------------------------------------------------------------------------
// Host side
// ---------------------------------------------------------------------------
static inline void launch_gemm(int ci, const float* W, const float* X, float* Y,
                               int co, hipStream_t stream) {
  int total = (co / 16) * (COLS / 64);
  int grid = (total + 7) / 8;      // 8 waves (256 threads) per block
  switch (ci) {
    case 8:    wmma_gemm_t<8><<<grid, 256, 0, stream>>>(W, X, Y, co); break;
    case 64:   wmma_gemm_t<64><<<grid, 256, 0, stream>>>(W, X, Y, co); break;
    case 128:  wmma_gemm_t<128><<<grid, 256, 0, stream>>>(W, X, Y, co); break;
    case 256:  wmma_gemm_t<256><<<grid, 256, 0, stream>>>(W, X, Y, co); break;
    case 2384: wmma_gemm_t<2384><<<grid, 256, 0, stream>>>(W, X, Y, co); break;
    default: break;
  }
}

static inline void launch_bn(const float* Y, float* mu, float* var,
                             const float* g, const float* b, float* dst,
                             int co, hipStream_t stream) {
  bn_stats_kernel<<<co, 256, 0, stream>>>(Y, mu, var);
  int grid = (int)(((size_t)co * COLS + 255) / 256);
  bn_apply_relu<<<grid, 256, 0, stream>>>(Y, mu, var, g, b, dst, co);
}

extern "C" void kernel_launch(void* const* d_in, const int* in_sizes, int n_in,
                              void* d_out, int out_size, void* d_ws, size_t ws_size,
                              hipStream_t stream) {
  (void)in_sizes; (void)n_in; (void)out_size; (void)ws_size;

  const float* in = (const float*)d_in[0];
  const float *pf_w[4], *pf_g[4], *pf_b[4];
  for (int i = 0; i < 4; ++i) {
    pf_w[i] = (const float*)d_in[1 + 3 * i];
    pf_g[i] = (const float*)d_in[2 + 3 * i];
    pf_b[i] = (const float*)d_in[3 + 3 * i];
  }
  const float *cls_w[3], *cls_g[3], *cls_b[3];
  for (int j = 0; j < 3; ++j) {
    cls_w[j] = (const float*)d_in[13 + 3 * j];
    cls_g[j] = (const float*)d_in[14 + 3 * j];
    cls_b[j] = (const float*)d_in[15 + 3 * j];
  }
  const float* cls3_w = (const float*)d_in[22];
  const float* cls3_bias = (const float*)d_in[23];
  float* out = (float*)d_out;

  // Workspace layout (floats)
  float* ws = (float*)d_ws;
  size_t o_feats = 0;                             // 8 * COLS
  size_t o_nrm   = o_feats + (size_t)8 * COLS;    // 3 * COLS
  size_t o_P     = o_nrm + (size_t)3 * COLS;      // 3 * COLS
  size_t o_cst   = o_P + (size_t)3 * COLS;        // 64
  size_t o_mu    = o_cst + 64;                    // 1024
  size_t o_var   = o_mu + 1024;                   // 1024
  size_t o_wp0   = o_var + 1024;                  // 64 * 8   (pf0 padded)
  size_t o_wp3   = o_wp0 + 64 * 8;                // 64 * 128 (cls3 padded)
  size_t o_bp3   = o_wp3 + 64 * 128;              // 64      (cls3 bias padded)
  size_t o_xcat  = o_bp3 + 64;                    // 2384 * COLS
  size_t o_y     = o_xcat + (size_t)2384 * COLS;  // 1024 * COLS
  size_t o_pa    = o_y + (size_t)1024 * COLS;     // 256 * COLS
  size_t o_pb    = o_pa + (size_t)256 * COLS;     // 256 * COLS

  float* feats = ws + o_feats;
  float* nrm   = ws + o_nrm;
  float* P     = ws + o_P;
  float* cstat = ws + o_cst;
  float* mu    = ws + o_mu;
  float* var   = ws + o_var;
  float* wpad0 = ws + o_wp0;
  float* wpad3 = ws + o_wp3;
  float* bpad3 = ws + o_bp3;
  float* xcat  = ws + o_xcat;
  float* ybuf  = ws + o_y;
  float* pbufA = ws + o_pa;
  float* pbufB = ws + o_pb;

  // xcat rows: [one_hot 0..15][f1 16..79][f2 80..207][f3 208..335]
  //            [f4 336..1359][maxpool 1360..2383]
  float* x_f1 = xcat + (size_t)16 * COLS;
  float* x_f2 = xcat + (size_t)80 * COLS;
  float* x_f3 = xcat + (size_t)208 * COLS;
  float* x_f4 = xcat + (size_t)336 * COLS;
  float* x_mx = xcat + (size_t)1360 * COLS;

  // --- weight padding (removes all guards from WMMA inner loops) ---
  pad_weight_kernel<<<(64 * 8 + 255) / 256, 256, 0, stream>>>(pf_w[0], wpad0, 64, 5, 64, 8);
  pad_weight_kernel<<<(64 * 128 + 255) / 256, 256, 0, stream>>>(cls3_w, wpad3, 50, 128, 64, 128);
  pad_weight_kernel<<<1, 256, 0, stream>>>(cls3_bias, bpad3, 50, 1, 64, 1);

  // --- geometry ---
  centroid_kernel<<<BATCH, 256, 0, stream>>>(in, nrm, cstat);
  ppf_kernel<<<COLS / 256, 256, 0, stream>>>(in, nrm, cstat, feats, P);
  alpha_kernel<<<COLS, 256, 0, stream>>>(P, feats);
  onehot_kernel<<<(16 * COLS) / 256, 256, 0, stream>>>(in, xcat);

  // --- point-feature MLP stack ---
  launch_gemm(8, wpad0, feats, ybuf, 64, stream);
  launch_bn(ybuf, mu, var, pf_g[0], pf_b[0], x_f1, 64, stream);

  launch_gemm(64, pf_w[1], x_f1, ybuf, 128, stream);
  launch_bn(ybuf, mu, var, pf_g[1], pf_b[1], x_f2, 128, stream);

  launch_gemm(128, pf_w[2], x_f2, ybuf, 128, stream);
  launch_bn(ybuf, mu, var, pf_g[2], pf_b[2], x_f3, 128, stream);

  launch_gemm(128, pf_w[3], x_f3, ybuf, 1024, stream);
  launch_bn(ybuf, mu, var, pf_g[3], pf_b[3], x_f4, 1024, stream);

  maxpool_bcast<<<1024 * BATCH, 256, 0, stream>>>(x_f4, x_mx);

  // --- classifier stack ---
  launch_gemm(2384, cls_w[0], xcat, ybuf, 256, stream);
  launch_bn(ybuf, mu, var, cls_g[0], cls_b[0], pbufA, 256, stream);

  launch_gemm(256, cls_w[1], pbufA, ybuf, 256, stream);
  launch_bn(ybuf, mu, var, cls_g[1], cls_b[1], pbufB, 256, stream);

  launch_gemm(256, cls_w[2], pbufB, ybuf, 128, stream);
  launch_bn(ybuf, mu, var, cls_g[2], cls_b[2], pbufA, 128, stream);

  // --- final linear 128 -> 50 ---
  {
    int total = 4 * (COLS / 64);
    int grid = (total + 7) / 8;
    wmma_gemm_final<<<grid, 256, 0, stream>>>(wpad3, bpad3, pbufA, out);
  }
}